// get_loss_13039520710859
// MI455X (gfx1250) — compile-verified
//
#include <hip/hip_runtime.h>
#include <math.h>

typedef __attribute__((ext_vector_type(2))) float v2f;
typedef __attribute__((ext_vector_type(8))) float v8f;

#define BATCH 16
#define CH    128
#define NPTS  65536
#define TREF  8            // refine_times (static in reference)
#define NCLS  40
#define MASKW (NPTS / 32)  // 2048 mask words per sample
#define WEIGHT 0.001f

// ---------------------------------------------------------------------------
// Kernel 0: zero the cumulative used-point bitmask (re-run every call so the
// graph replay is deterministic).
// ---------------------------------------------------------------------------
__global__ void k_zero_mask(unsigned* __restrict__ mask) {
    int i = blockIdx.x * 256 + threadIdx.x;
    if (i < BATCH * MASKW) mask[i] = 0u;
}

// ---------------------------------------------------------------------------
// Kernel A: per-(b,c) masked max/argmax over N.  One workgroup per (b,c).
// Streams 256 KB of contiguous f32 per block as float4 (global_load_b128),
// mask bits served from LDS, next tile prefetched.  Deterministic tree
// reduction with first-occurrence (lowest index) tie-break like jnp.argmax.
// ---------------------------------------------------------------------------
__global__ __launch_bounds__(256) void k_masked_argmax(
        const float* __restrict__ features,
        const unsigned* __restrict__ mask,
        float* __restrict__ feat_out,   // [BATCH][CH][TREF]
        int* __restrict__ picks,        // [BATCH*CH]
        int t) {
    __shared__ unsigned smask[MASKW];   // 8 KB: this sample's mask
    __shared__ float sval[256];
    __shared__ int   sidx[256];

    const int bc  = blockIdx.x;         // b*CH + c
    const int b   = bc >> 7;
    const int tid = threadIdx.x;

    for (int i = tid; i < MASKW; i += 256) smask[i] = mask[b * MASKW + i];
    __syncthreads();

    const float4* fp = (const float4*)(features + ((size_t)bc << 16));
    float best    = -INFINITY;
    int   bestidx = 0;

    constexpr int ITERS = NPTS / (256 * 4);   // 64
    for (int it = 0; it < ITERS; ++it) {
        const int vec = it * 256 + tid;       // float4 index
        if (it + 1 < ITERS)
            __builtin_prefetch(&fp[vec + 256], 0, 1);   // global_prefetch
        float4 v = fp[vec];
        const int      n0 = vec << 2;
        const unsigned mw = smask[n0 >> 5];
        const unsigned sh = (unsigned)(n0 & 31);
        if (!((mw >> (sh + 0u)) & 1u) && v.x > best) { best = v.x; bestidx = n0 + 0; }
        if (!((mw >> (sh + 1u)) & 1u) && v.y > best) { best = v.y; bestidx = n0 + 1; }
        if (!((mw >> (sh + 2u)) & 1u) && v.z > best) { best = v.z; bestidx = n0 + 2; }
        if (!((mw >> (sh + 3u)) & 1u) && v.w > best) { best = v.w; bestidx = n0 + 3; }
    }

    sval[tid] = best; sidx[tid] = bestidx;
    __syncthreads();
    for (int s = 128; s > 0; s >>= 1) {
        if (tid < s) {
            float ov = sval[tid + s]; int oi = sidx[tid + s];
            if (ov > sval[tid] || (ov == sval[tid] && oi < sidx[tid])) {
                sval[tid] = ov; sidx[tid] = oi;
            }
        }
        __syncthreads();
    }
    if (tid == 0) {
        feat_out[bc * TREF + t] = sval[0];
        picks[bc] = sidx[0];
    }
}

// ---------------------------------------------------------------------------
// Kernel B: fold this round's 2048 argmax picks into the cumulative mask.
// Separate kernel so no block of round t can observe round t's picks.
// ---------------------------------------------------------------------------
__global__ void k_apply_picks(const int* __restrict__ picks,
                              unsigned* __restrict__ mask) {
    int i = blockIdx.x * 256 + threadIdx.x;
    if (i < BATCH * CH) {
        int b   = i >> 7;
        int idx = picks[i];
        atomicOr(&mask[b * MASKW + (idx >> 5)], 1u << (idx & 31));
    }
}

// ---------------------------------------------------------------------------
// 8x8 symmetric eigensolve (cyclic Jacobi); returns sum of sqrt(eigenvalues)
// = nuclear norm of the original 128x8 matrix.
// ---------------------------------------------------------------------------
__device__ inline float jacobi_nuclear8(float G[TREF][TREF]) {
    for (int sweep = 0; sweep < 15; ++sweep) {
        for (int p = 0; p < TREF - 1; ++p) {
            for (int q = p + 1; q < TREF; ++q) {
                float apq = G[p][q];
                if (fabsf(apq) < 1e-10f) continue;
                float app = G[p][p], aqq = G[q][q];
                float tau = (aqq - app) / (2.0f * apq);
                float rt  = sqrtf(1.0f + tau * tau);
                float tv  = (tau >= 0.0f) ? 1.0f / (tau + rt) : 1.0f / (tau - rt);
                float cth = 1.0f / sqrtf(1.0f + tv * tv);
                float sth = tv * cth;
                for (int k = 0; k < TREF; ++k) {
                    float gkp = G[k][p], gkq = G[k][q];
                    G[k][p] = cth * gkp - sth * gkq;
                    G[k][q] = sth * gkp + cth * gkq;
                }
                for (int k = 0; k < TREF; ++k) {
                    float gpk = G[p][k], gqk = G[q][k];
                    G[p][k] = cth * gpk - sth * gqk;
                    G[q][k] = sth * gpk + cth * gqk;
                }
            }
        }
    }
    float s = 0.0f;
    for (int i = 0; i < TREF; ++i) s += sqrtf(fmaxf(G[i][i], 0.0f));
    return s;
}

// ---------------------------------------------------------------------------
// Kernel C: one wave per batch.  Gram matrix G = F^T F (F = [128 x 8]) via
// V_WMMA_F32_16X16X4_F32 accumulating K=128 in 32 chunks of 4 (A == B since
// G is symmetric), then Jacobi eigensolve + NLL + final scalar.
// ---------------------------------------------------------------------------
__global__ __launch_bounds__(512) void k_nuclear_nll(
        const float* __restrict__ feat,    // [BATCH][CH][TREF]
        const float* __restrict__ pred,    // [BATCH][NCLS] (log-softmax)
        const int*  __restrict__ target,   // [BATCH]
        float* __restrict__ out) {
    __shared__ float Gs[BATCH][16][16];    // 16 KB
    __shared__ float nuc[BATCH];

    const int wave  = threadIdx.x >> 5;    // batch index, 16 waves
    const int lane  = threadIdx.x & 31;
    const int m     = lane & 15;           // A-matrix row / C-matrix column
    const int khalf = (lane >> 4) << 1;    // K pair: lanes 0-15 -> {0,1}, 16-31 -> {2,3}

    const float* fb = feat + wave * (CH * TREF);
    const float  mval = (m < TREF) ? 1.0f : 0.0f;  // zero-pad rows 8..15

    v8f acc = {};
    for (int kc = 0; kc < CH / 4; ++kc) {
        const int c0 = kc * 4 + khalf;
        v2f a;
        a.x = fb[c0       * TREF + (m & 7)] * mval;   // A[m][k]   = F[c0  ][m]
        a.y = fb[(c0 + 1) * TREF + (m & 7)] * mval;   // A[m][k+1] = F[c0+1][m]
        // G += A * A^T  (A and B fragments identical for the Gram matrix)
        acc = __builtin_amdgcn_wmma_f32_16x16x4_f32(
                  false, a, false, a, (short)0, acc, false, false);
    }

    // C/D layout: VGPR r -> lanes 0-15: (M=r, N=lane), lanes 16-31: (M=r+8, N=lane-16)
    const int mrow = (lane < 16) ? 0 : 8;
#pragma unroll
    for (int r = 0; r < 8; ++r)
        Gs[wave][mrow + r][m] = acc[r];
    __syncthreads();

    if (lane == 0) {
        float G[TREF][TREF];
        for (int i = 0; i < TREF; ++i)
            for (int j = 0; j < TREF; ++j)
                G[i][j] = Gs[wave][i][j];
        nuc[wave] = jacobi_nuclear8(G);
    }
    __syncthreads();

    if (threadIdx.x == 0) {
        float s = 0.0f;
        for (int b = 0; b < BATCH; ++b) s += nuc[b];
        float nuclear = s / (float)BATCH;
        float nll = 0.0f;
        for (int i = 0; i < BATCH; ++i) nll += pred[i * NCLS + target[i]];
        nll = -nll / (float)BATCH;
        out[0] = nll + nuclear * WEIGHT;
    }
}

// ---------------------------------------------------------------------------
extern "C" void kernel_launch(void* const* d_in, const int* in_sizes, int n_in,
                              void* d_out, int out_size, void* d_ws, size_t ws_size,
                              hipStream_t stream) {
    const float* pred     = (const float*)d_in[0];
    const int*   target   = (const int*)d_in[1];
    const float* features = (const float*)d_in[2];
    // d_in[3] = refine_times; static 8 in the reference (controls launch count)

    unsigned char* ws = (unsigned char*)d_ws;
    unsigned* mask = (unsigned*)ws;                                   // 131072 B
    int*      picks = (int*)(ws + (size_t)BATCH * MASKW * 4);         //   8192 B
    float*    feat  = (float*)(ws + (size_t)BATCH * MASKW * 4 + 8192);//  65536 B
    float*    out   = (float*)d_out;

    k_zero_mask<<<(BATCH * MASKW + 255) / 256, 256, 0, stream>>>(mask);

    for (int t = 0; t < TREF; ++t) {
        k_masked_argmax<<<BATCH * CH, 256, 0, stream>>>(features, mask, feat, picks, t);
        k_apply_picks<<<(BATCH * CH + 255) / 256, 256, 0, stream>>>(picks, mask);
    }

    k_nuclear_nll<<<1, 512, 0, stream>>>(feat, pred, target, out);
}